// Sampler_5257039970947
// MI455X (gfx1250) — compile-verified
//
#include <hip/hip_runtime.h>
#include <math.h>

// ---------------------------------------------------------------------------
// MI455X sampler pipeline (sort-free).
//  Kernel 1: penalties + temperature + row max        (1 read + 1 write of V)
//  Kernel 2: radix-select thresholds (top-k / top-p)  (~5 L2-resident reads)
//            + min-p, final masking in place          (1 read + 1 write)
//  All per-level 2048-bin suffix scans run on V_WMMA_F32_16X16X4_F32
//  (counts < 2^24 -> exact in f32).  Histograms via LDS ds_add_f32 atomics.
// ---------------------------------------------------------------------------

#define NSEQ   256
#define VOCAB  128000
#define PLEN   1024
#define OLEN   512
#define CHUNK  12800      // 10 chunks of the vocab per row in kernel 1
#define T1     1024
#define T2     1024
#define NBINS  2048

typedef float v2f __attribute__((ext_vector_type(2)));
typedef float v8f __attribute__((ext_vector_type(8)));

// monotone float<->uint key (ascending key order == ascending float order)
__device__ __forceinline__ unsigned f2key(float v) {
  unsigned b = __float_as_uint(v);
  return b ^ ((b & 0x80000000u) ? 0xFFFFFFFFu : 0x80000000u);
}
__device__ __forceinline__ float key2f(unsigned k) {
  unsigned b = (k & 0x80000000u) ? (k ^ 0x80000000u) : ~k;
  return __uint_as_float(b);
}

// ---------------------------------------------------------------------------
// Within-16 suffix sums of one 16x16 tile (16 segments x 16 bins) of a 2048-
// entry LDS array, as D = A x U with U[j][i] = (j >= i), chained over 4 K=4
// WMMA steps.  Exact for integral counts in f32.  Whole wave must be active.
//   A 16x4 layout : lanes 0-15 rows M, {vgpr0,vgpr1}={K0,K1}; lanes16-31 {K2,K3}
//   B 4x16 layout : lanes 0-15 cols N, rows {K0,K1};          lanes16-31 {K2,K3}
//   D 16x16       : vgpr r -> row r (lanes 0-15) / row r+8 (lanes 16-31)
// ---------------------------------------------------------------------------
__device__ __forceinline__ void wmma_tile_suffix16(const float* __restrict__ src,
                                                   float* __restrict__ dst,
                                                   int tile, int lane) {
  const float* base = src + tile * 256;
  const int m  = lane & 15;
  const int hi = lane >> 4;
  v8f d = {0.f, 0.f, 0.f, 0.f, 0.f, 0.f, 0.f, 0.f};
#pragma unroll
  for (int kc = 0; kc < 4; ++kc) {
    const int k0 = kc * 4 + hi * 2;
    v2f a, b;
    a.x = base[m * 16 + k0 + 0];
    a.y = base[m * 16 + k0 + 1];
    b.x = (k0 + 0 >= m) ? 1.0f : 0.0f;   // U[k0+0][m]
    b.y = (k0 + 1 >= m) ? 1.0f : 0.0f;   // U[k0+1][m]
    d = __builtin_amdgcn_wmma_f32_16x16x4_f32(false, a, false, b,
                                              (short)0, d, false, false);
  }
#pragma unroll
  for (int r = 0; r < 8; ++r)
    dst[tile * 256 + (hi * 8 + r) * 16 + m] = d[r];
}

// ---------------------------------------------------------------------------
// Kernel 1: penalties + temperature, one workgroup per row.
// Token lists (1536 ids) live in LDS; vocab processed in 12800-wide chunks
// with a u32 LDS histogram: hi 16 bits = prompt hits, lo 16 = output count.
// ---------------------------------------------------------------------------
__global__ void __launch_bounds__(T1)
penalize_kernel(const float* __restrict__ logits,
                const float* __restrict__ pres, const float* __restrict__ freq,
                const float* __restrict__ rep,  const float* __restrict__ temp,
                const int* __restrict__ ptoks,  const int* __restrict__ otoks,
                float* __restrict__ out, float* __restrict__ rowMax) {
  __shared__ unsigned s_hist[CHUNK];
  __shared__ int s_pt[PLEN];
  __shared__ int s_ot[OLEN];
  __shared__ float s_red[32];

  const int row = blockIdx.x;
  const int tid = threadIdx.x;
  const size_t roff = (size_t)row * VOCAB;

  for (int i = tid; i < PLEN; i += T1) s_pt[i] = ptoks[row * PLEN + i];
  for (int i = tid; i < OLEN; i += T1) s_ot[i] = otoks[row * OLEN + i];

  const float rp = rep[row], fp = freq[row], pp = pres[row];
  const float invt = 1.0f / temp[row];
  float lmax = -__builtin_inff();
  __syncthreads();

  for (int cbase = 0; cbase < VOCAB; cbase += CHUNK) {
    for (int i = tid; i < CHUNK; i += T1) s_hist[i] = 0u;
    __syncthreads();
    for (int i = tid; i < PLEN; i += T1) {
      unsigned off = (unsigned)(s_pt[i] - cbase);
      if (off < (unsigned)CHUNK) atomicAdd(&s_hist[off], 0x10000u);
    }
    for (int i = tid; i < OLEN; i += T1) {
      unsigned off = (unsigned)(s_ot[i] - cbase);
      if (off < (unsigned)CHUNK) atomicAdd(&s_hist[off], 1u);
    }
    __syncthreads();
    for (int i = tid; i < CHUNK; i += T1) {
      const size_t gi = roff + (size_t)(cbase + i);
      __builtin_prefetch(&logits[gi + T1], 0, 0);
      float l = logits[gi];
      const unsigned h  = s_hist[i];
      const unsigned oc = h & 0xFFFFu;
      if (h != 0u) l = (l > 0.0f) ? (l / rp) : (l * rp);
      l -= fp * (float)oc;
      if (oc != 0u) l -= pp;
      l *= invt;
      out[gi] = l;
      lmax = fmaxf(lmax, l);
    }
    __syncthreads();
  }

  // row max reduction (wave32)
  for (int o = 16; o > 0; o >>= 1) lmax = fmaxf(lmax, __shfl_xor(lmax, o, 32));
  if ((tid & 31) == 0) s_red[tid >> 5] = lmax;
  __syncthreads();
  if (tid < 32) {
    float v = s_red[tid];
    for (int o = 16; o > 0; o >>= 1) v = fmaxf(v, __shfl_xor(v, o, 32));
    if (tid == 0) rowMax[row] = v;
  }
}

// ---------------------------------------------------------------------------
// Kernel 2: radix-select thresholds + final mask.  One workgroup per row.
// ---------------------------------------------------------------------------
__global__ void __launch_bounds__(T2)
select_mask_kernel(const float* __restrict__ topp, const float* __restrict__ minp,
                   const int* __restrict__ topk, const float* __restrict__ rowMax,
                   float* __restrict__ lg) {
  __shared__ float hcnt[NBINS];
  __shared__ float hwgt[NBINS];
  __shared__ float scnt[NBINS + 16];
  __shared__ float swgt[NBINS + 16];
  __shared__ float l0w [NBINS + 16];
  __shared__ float segA[128];
  __shared__ float segB[128];
  __shared__ int   s_sel;

  const int row  = blockIdx.x;
  const int tid  = threadIdx.x;
  const int wave = tid >> 5, lane = tid & 31;
  const size_t roff = (size_t)row * VOCAB;

  const float m = rowMax[row];
  int k = topk[row]; if (k < 1) k = 1; if (k > VOCAB) k = VOCAB;
  const float kf = (float)k;
  const float tp = topp[row];
  const float mp = minp[row];

  auto zero_hists = [&]() {
    for (int i = tid; i < NBINS; i += T2) { hcnt[i] = 0.0f; hwgt[i] = 0.0f; }
  };
  // hist -> full suffix sums (scnt/swgt), sentinel at [NBINS]=0
  auto suffixes = [&]() {
    __syncthreads();
    if (wave < 8)       wmma_tile_suffix16(hcnt, scnt, wave,     lane);
    else if (wave < 16) wmma_tile_suffix16(hwgt, swgt, wave - 8, lane);
    __syncthreads();
    if (tid == 0)  { float a = 0.f; for (int s = 127; s >= 0; --s) { segA[s] = a; a += scnt[s * 16]; } }
    if (tid == 32) { float a = 0.f; for (int s = 127; s >= 0; --s) { segB[s] = a; a += swgt[s * 16]; } }
    __syncthreads();
    for (int i = tid; i < NBINS; i += T2) { scnt[i] += segA[i >> 4]; swgt[i] += segB[i >> 4]; }
    if (tid == 0) { scnt[NBINS] = 0.0f; swgt[NBINS] = 0.0f; }
    __syncthreads();
  };
  // find b : base+suf[b+1] < target <= base+suf[b]   (suffix is non-increasing)
  auto select_from = [&](const float* suf, float base, float target) -> int {
    if (tid == 0) s_sel = 0;
    __syncthreads();
    for (int b = tid; b < NBINS; b += T2)
      if (base + suf[b + 1] < target && base + suf[b] >= target) s_sel = b;
    __syncthreads();
    return s_sel;
  };

  // ---- Phase A: level-0 histogram over all tokens ----
  zero_hists();
  __syncthreads();
  for (int i = tid; i < VOCAB; i += T2) {
    __builtin_prefetch(&lg[roff + i + 4 * T2], 0, 0);
    const float v = lg[roff + i];
    const unsigned key = f2key(v);
    atomicAdd(&hcnt[key >> 21], 1.0f);
    atomicAdd(&hwgt[key >> 21], __expf(v - m));
  }
  suffixes();
  for (int i = tid; i <= NBINS; i += T2) l0w[i] = swgt[i];  // save for top-p L0
  __syncthreads();
  const int b0k = select_from(scnt, 0.0f, kf);
  float kA = scnt[b0k + 1];
  float wA = swgt[b0k + 1];
  __syncthreads();

  // ---- Phase B: top-k level 1 ----
  zero_hists();
  __syncthreads();
  {
    const unsigned pre = (unsigned)b0k;
    for (int i = tid; i < VOCAB; i += T2) {
      const float v = lg[roff + i];
      const unsigned key = f2key(v);
      if ((key >> 21) == pre) {
        const int b = (key >> 10) & 2047;
        atomicAdd(&hcnt[b], 1.0f);
        atomicAdd(&hwgt[b], __expf(v - m));
      }
    }
  }
  suffixes();
  const int b1k = select_from(scnt, kA, kf);
  kA += scnt[b1k + 1];
  wA += swgt[b1k + 1];
  __syncthreads();

  // ---- Phase C: top-k level 2 (exact key) ----
  zero_hists();
  __syncthreads();
  {
    const unsigned pre = ((unsigned)b0k << 11) | (unsigned)b1k;
    for (int i = tid; i < VOCAB; i += T2) {
      const float v = lg[roff + i];
      const unsigned key = f2key(v);
      if ((key >> 10) == pre) {
        const int b = key & 1023;
        atomicAdd(&hcnt[b], 1.0f);
        atomicAdd(&hwgt[b], __expf(v - m));
      }
    }
  }
  suffixes();
  const int b2k = select_from(scnt, kA, kf);
  const float kthVal = key2f(((unsigned)b0k << 21) | ((unsigned)b1k << 10) | (unsigned)b2k);
  const float Zp = wA + swgt[b2k];        // softmax mass of the top-k keep set
  const float P  = tp * Zp;               // keep v iff mass strictly above < P
  __syncthreads();

  // ---- Phase D: top-p level 0 (reuse saved L0 weight suffix, no scan) ----
  const int bp0 = select_from(l0w, 0.0f, P);
  float wAp = l0w[bp0 + 1];
  __syncthreads();

  // ---- Phase E: top-p level 1 ----
  zero_hists();
  __syncthreads();
  {
    const unsigned pre = (unsigned)bp0;
    for (int i = tid; i < VOCAB; i += T2) {
      const float v = lg[roff + i];
      const unsigned key = f2key(v);
      if ((key >> 21) == pre)
        atomicAdd(&hwgt[(key >> 10) & 2047], __expf(v - m));
    }
  }
  suffixes();
  const int bp1 = select_from(swgt, wAp, P);
  wAp += swgt[bp1 + 1];
  __syncthreads();

  // ---- Phase F: top-p level 2 (exact key) ----
  zero_hists();
  __syncthreads();
  {
    const unsigned pre = ((unsigned)bp0 << 11) | (unsigned)bp1;
    for (int i = tid; i < VOCAB; i += T2) {
      const float v = lg[roff + i];
      const unsigned key = f2key(v);
      if ((key >> 10) == pre)
        atomicAdd(&hwgt[key & 1023], __expf(v - m));
    }
  }
  suffixes();
  const int bp2 = select_from(swgt, wAp, P);
  const float tpVal = key2f(((unsigned)bp0 << 21) | ((unsigned)bp1 << 10) | (unsigned)bp2);

  // ---- Phase G: final masking (top-k, top-p, min-p are all value thresholds)
  float thr = fmaxf(kthVal, tpVal);
  thr = fmaxf(thr, m + __logf(mp));       // min_p: keep iff exp(v-m) >= min_p
  __syncthreads();
  for (int i = tid; i < VOCAB; i += T2) {
    const float v = lg[roff + i];
    lg[roff + i] = (v >= thr) ? v : -__builtin_inff();
  }
}

// ---------------------------------------------------------------------------
extern "C" void kernel_launch(void* const* d_in, const int* in_sizes, int n_in,
                              void* d_out, int out_size, void* d_ws, size_t ws_size,
                              hipStream_t stream) {
  (void)in_sizes; (void)n_in; (void)out_size; (void)ws_size;
  const float* logits = (const float*)d_in[0];
  const float* pres   = (const float*)d_in[1];
  const float* freq   = (const float*)d_in[2];
  const float* rep    = (const float*)d_in[3];
  const float* temp   = (const float*)d_in[4];
  const float* topp   = (const float*)d_in[5];
  const float* minp   = (const float*)d_in[6];
  const int*   ptoks  = (const int*)d_in[7];
  const int*   otoks  = (const int*)d_in[8];
  const int*   topk   = (const int*)d_in[9];
  float* out    = (float*)d_out;
  float* rowMax = (float*)d_ws;   // 256 floats of scratch

  penalize_kernel<<<NSEQ, T1, 0, stream>>>(logits, pres, freq, rep, temp,
                                           ptoks, otoks, out, rowMax);
  select_mask_kernel<<<NSEQ, T2, 0, stream>>>(topp, minp, topk, rowMax, out);
}